// Model_19069654794449
// MI455X (gfx1250) — compile-verified
//
#include <hip/hip_runtime.h>
#include <cmath>

#ifndef M_PI
#define M_PI 3.14159265358979323846
#endif

typedef __attribute__((ext_vector_type(2))) float v2f;
typedef __attribute__((ext_vector_type(8))) float v8f;
typedef __attribute__((ext_vector_type(4))) unsigned int u32x4;
typedef __attribute__((ext_vector_type(4))) int i32x4;
typedef __attribute__((ext_vector_type(8))) int i32x8;

#define NVERT 642
#define NFACE 512
#define NPIX  65536
#define TSZ   4
#define FTOT  (13 * NFACE)   // 6656 floats of SoA face data

// workspace layout (float offsets)
#define WS_VTX  0            // 642*3 = 1926 floats
#define WS_FACE 2048         // 13 SoA arrays * 512 = 6656 floats
#define WS_PART 8704         // 4096 per-wave partials
#define NWAVE   4096

// -------------------------------------------------------------------------
// Kernel 1: camera transform  v = R @ (vert - eye)
// -------------------------------------------------------------------------
__global__ void vtx_kernel(const float* __restrict__ verts, float* __restrict__ ws,
                           float ex, float ey, float ez,
                           float r00, float r01, float r02,
                           float r10, float r11, float r12,
                           float r20, float r21, float r22) {
    int i = blockIdx.x * blockDim.x + threadIdx.x;
    if (i >= NVERT) return;
    float vx = verts[i * 3 + 0] - ex;
    float vy = verts[i * 3 + 1] - ey;
    float vz = verts[i * 3 + 2] - ez;
    float* o = ws + WS_VTX + i * 3;
    o[0] = r00 * vx + r01 * vy + r02 * vz;
    o[1] = r10 * vx + r11 * vy + r12 * vz;
    o[2] = r20 * vx + r21 * vy + r22 * vz;
}

// -------------------------------------------------------------------------
// Kernel 2: per-face edge-function setup (divided affine coefficients).
// SoA rows: [A0 B0 C0 A1 B1 C1 A2 B2 C2 Z0 Z1 Z2 ND], each [NFACE]
// -------------------------------------------------------------------------
__global__ void face_kernel(const int* __restrict__ faces, float* __restrict__ ws) {
    int f = blockIdx.x * blockDim.x + threadIdx.x;
    if (f >= NFACE) return;
    const float* v = ws + WS_VTX;
    int i0 = faces[f * 3 + 0], i1 = faces[f * 3 + 1], i2 = faces[f * 3 + 2];
    float ax = v[i0 * 3], ay = v[i0 * 3 + 1], az = v[i0 * 3 + 2];
    float bx = v[i1 * 3], by = v[i1 * 3 + 1], bz = v[i1 * 3 + 2];
    float cx = v[i2 * 3], cy = v[i2 * 3 + 1], cz = v[i2 * 3 + 2];

    float denom = (bx - ax) * (cy - ay) - (by - ay) * (cx - ax);
    float nd = (fabsf(denom) > 1e-8f) ? 1.0f : 0.0f;
    float d  = (nd != 0.0f) ? denom : 1.0f;
    float inv = 1.0f / d;

    float* F = ws + WS_FACE;
    F[0 * NFACE + f] = -(cy - by) * inv;
    F[1 * NFACE + f] =  (cx - bx) * inv;
    F[2 * NFACE + f] = ((cy - by) * bx - (cx - bx) * by) * inv;
    F[3 * NFACE + f] = -(ay - cy) * inv;
    F[4 * NFACE + f] =  (ax - cx) * inv;
    F[5 * NFACE + f] = ((ay - cy) * cx - (ax - cx) * cy) * inv;
    F[6 * NFACE + f] = -(by - ay) * inv;
    F[7 * NFACE + f] =  (bx - ax) * inv;
    F[8 * NFACE + f] = ((by - ay) * ax - (bx - ax) * ay) * inv;
    F[9  * NFACE + f] = az;
    F[10 * NFACE + f] = bz;
    F[11 * NFACE + f] = cz;
    F[12 * NFACE + f] = nd;
}

// -------------------------------------------------------------------------
// Kernel 3: fused rasterize + z-argmin + texture + squared error.
// Face table staged global->LDS once per block via the Tensor Data Mover;
// edge functions per 16-pixel x 16-face tile via V_WMMA_F32_16X16X4_F32.
// -------------------------------------------------------------------------
__global__ void __launch_bounds__(256)
raster_kernel(const float* __restrict__ Fb, const float* __restrict__ tex,
              const float* __restrict__ ref, float* __restrict__ part) {
    __shared__ float fs[FTOT];

    // ---- TDM: DMA the whole 26KB SoA face table into LDS (wave 0 only) ----
    if ((threadIdx.x >> 5) == 0) {
        unsigned long long ga = (unsigned long long)(uintptr_t)Fb;   // global addr
        unsigned int la = (unsigned int)(uintptr_t)&fs[0];           // LDS offset = addr[31:0]

        u32x4 g0;
        g0[0] = 1u;                                                  // count=1, no gather
        g0[1] = la;                                                  // lds_addr
        g0[2] = (unsigned int)ga;                                    // global_addr[31:0]
        g0[3] = ((unsigned int)(ga >> 32) & 0x01FFFFFFu) | 0x80000000u; // ga[56:32], type=2

        i32x8 g1;
        g1[0] = 0x00020000;                    // workgroup_mask=0, data_size=4B
        g1[1] = (int)((FTOT & 0xFFFF) << 16);  // tensor_dim0[15:0]
        g1[2] = (int)((FTOT >> 16) | (1 << 16)); // tensor_dim0[31:16] | tensor_dim1=1
        g1[3] = (int)((FTOT & 0xFFFF) << 16);  // tile_dim0 = FTOT
        g1[4] = 1;                             // tile_dim1=1, tile_dim2=0
        g1[5] = FTOT;                          // tensor_dim0_stride[31:0]
        g1[6] = (int)((FTOT & 0xFFFF) << 16);  // stride0[47:32]=0 | stride1[15:0]
        g1[7] = 0;

        i32x4 z4 = {0, 0, 0, 0};
#if defined(__clang_major__) && (__clang_major__ >= 23)
        i32x8 z8 = {0, 0, 0, 0, 0, 0, 0, 0};
        __builtin_amdgcn_tensor_load_to_lds(g0, g1, z4, z4, z8, 0);
#else
        __builtin_amdgcn_tensor_load_to_lds(g0, g1, z4, z4, 0);
#endif
        __builtin_amdgcn_s_wait_tensorcnt(0);
    }
    __syncthreads();

    const int lane = threadIdx.x & 31;
    const int wave = blockIdx.x * (blockDim.x >> 5) + (threadIdx.x >> 5);
    const int l16  = lane & 15;
    const int half = lane >> 4;
    const int pixBase = wave * 16;

    // prefetch the reference-image lines this tile will need at the end
    __builtin_prefetch(ref + 0 * NPIX + pixBase, 0, 3);
    __builtin_prefetch(ref + 1 * NPIX + pixBase, 0, 3);
    __builtin_prefetch(ref + 2 * NPIX + pixBase, 0, 3);

    // A operand: lane L<16 -> {px,py} (K=0,1); lane 16+L -> {1,0} (K=2,3)
    const int pA = pixBase + l16;
    const float px = ((float)(pA & 255) + 0.5f) * (2.0f / 256.0f) - 1.0f;
    const float py = 1.0f - ((float)(pA >> 8) + 0.5f) * (2.0f / 256.0f);
    v2f A;
    A.x = half ? 1.0f : px;
    A.y = half ? 0.0f : py;

    const float INF = __builtin_inff();
    float bd[8];
    int   bi[8];
#pragma unroll
    for (int r = 0; r < 8; ++r) { bd[r] = INF; bi[r] = 0; }

    for (int g = 0; g < NFACE / 16; ++g) {
        const int face = g * 16 + l16;   // column N = l16 (both halves)

        // branchless B-operand fetch from LDS:
        //   half 0: {A_e, B_e} rows; half 1: {C_e, 0} rows
        float b0x = fs[(half ? 2 * NFACE : 0 * NFACE) + face];
        float b0y = fs[1 * NFACE + face];
        float b1x = fs[(half ? 5 * NFACE : 3 * NFACE) + face];
        float b1y = fs[4 * NFACE + face];
        float b2x = fs[(half ? 8 * NFACE : 6 * NFACE) + face];
        float b2y = fs[7 * NFACE + face];
        v2f B0, B1, B2;
        B0.x = b0x; B0.y = half ? 0.0f : b0y;
        B1.x = b1x; B1.y = half ? 0.0f : b1y;
        B2.x = b2x; B2.y = half ? 0.0f : b2y;

        const float z0 = fs[9  * NFACE + face];
        const float z1 = fs[10 * NFACE + face];
        const float z2 = fs[11 * NFACE + face];
        const bool  nd = fs[12 * NFACE + face] != 0.0f;

        const v8f cz = {};
        v8f w0v = __builtin_amdgcn_wmma_f32_16x16x4_f32(false, A, false, B0, (short)0, cz, false, false);
        v8f w1v = __builtin_amdgcn_wmma_f32_16x16x4_f32(false, A, false, B1, (short)0, cz, false, false);
        v8f w2v = __builtin_amdgcn_wmma_f32_16x16x4_f32(false, A, false, B2, (short)0, cz, false, false);

#pragma unroll
        for (int r = 0; r < 8; ++r) {
            float w0 = w0v[r], w1 = w1v[r], w2 = w2v[r];
            float depth = fmaf(w0, z0, fmaf(w1, z1, w2 * z2));
            bool vis = nd & (w0 >= 0.0f) & (w1 >= 0.0f) & (w2 >= 0.0f) & (depth > 0.0f);
            float zb = vis ? depth : INF;
            bool take = zb < bd[r];          // strict < keeps earliest face (argmin)
            bd[r] = take ? zb   : bd[r];
            bi[r] = take ? face : bi[r];
        }
    }

    // cross-lane argmin over the 16 face-columns, lexicographic (depth, idx)
    float myD = INF;
    int   myI = 0;
#pragma unroll
    for (int r = 0; r < 8; ++r) {
        float d = bd[r];
        int   i = bi[r];
#pragma unroll
        for (int m = 1; m < 16; m <<= 1) {
            float od = __shfl_xor(d, m, 32);
            int   oi = __shfl_xor(i, m, 32);
            bool  t  = (od < d) || ((od == d) && (oi < i));
            d = t ? od : d;
            i = t ? oi : i;
        }
        // lane j (0..15) owns pixel j = (j&7) + 8*(j>>3); half-1 rows come from lane j+8
        int src = lane + ((lane & 8) ? 8 : 0);
        d = __shfl(d, src, 32);
        i = __shfl(i, src, 32);
        if ((lane & 7) == r) { myD = d; myI = i; }
    }

    float err = 0.0f;
    if (lane < 16) {
        const int pp = pixBase + lane;
        const bool hit = (myD < INF);
        // recompute winning barycentrics from the LDS coefficients (affine in p)
        const float qx = ((float)(pp & 255) + 0.5f) * (2.0f / 256.0f) - 1.0f;
        const float qy = 1.0f - ((float)(pp >> 8) + 0.5f) * (2.0f / 256.0f);
        float w0 = fmaf(fs[0 * NFACE + myI], qx, fmaf(fs[1 * NFACE + myI], qy, fs[2 * NFACE + myI]));
        float w1 = fmaf(fs[3 * NFACE + myI], qx, fmaf(fs[4 * NFACE + myI], qy, fs[5 * NFACE + myI]));
        float w2 = fmaf(fs[6 * NFACE + myI], qx, fmaf(fs[7 * NFACE + myI], qy, fs[8 * NFACE + myI]));
        int t0 = min(max((int)rintf(w0 * (float)(TSZ - 1)), 0), TSZ - 1);
        int t1 = min(max((int)rintf(w1 * (float)(TSZ - 1)), 0), TSZ - 1);
        int t2 = min(max((int)rintf(w2 * (float)(TSZ - 1)), 0), TSZ - 1);
        const float* tp = tex + (size_t)(((myI * TSZ + t0) * TSZ + t1) * TSZ + t2) * 3;
#pragma unroll
        for (int c = 0; c < 3; ++c) {
            float col = hit ? tanhf(tp[c]) : 0.0f;
            float dv  = col - ref[c * NPIX + pp];
            err = fmaf(dv, dv, err);
        }
    }
    // fixed-order intra-wave sum
#pragma unroll
    for (int m = 1; m < 32; m <<= 1) err += __shfl_xor(err, m, 32);
    if (lane == 0) part[wave] = err;
}

// -------------------------------------------------------------------------
// Kernel 4: deterministic final reduction of 4096 per-wave partials.
// -------------------------------------------------------------------------
__global__ void reduce_kernel(const float* __restrict__ part, float* __restrict__ out) {
    __shared__ float sm[256];
    int t = threadIdx.x;
    float s = 0.0f;
    for (int i = 0; i < NWAVE / 256; ++i) s += part[t + i * 256];
    sm[t] = s;
    __syncthreads();
    for (int w = 128; w > 0; w >>= 1) {
        if (t < w) sm[t] += sm[t + w];
        __syncthreads();
    }
    if (t == 0) out[0] = sm[0];
}

// -------------------------------------------------------------------------
extern "C" void kernel_launch(void* const* d_in, const int* in_sizes, int n_in,
                              void* d_out, int out_size, void* d_ws, size_t ws_size,
                              hipStream_t stream) {
    const float* verts = (const float*)d_in[0];  // (1, 642, 3)
    const float* tex   = (const float*)d_in[1];  // (1, 512, 4, 4, 4, 3)
    const float* ref   = (const float*)d_in[2];  // (1, 3, 256, 256)
    const int*   faces = (const int*)d_in[3];    // (1, 512, 3)
    float* ws = (float*)d_ws;

    // camera constants, computed in double to match the numpy construction
    const double az = M_PI / 2.0, el = 0.0, dist = 2.732;
    const double ex = dist * cos(el) * sin(az);
    const double ey = dist * sin(el);
    const double ez = -dist * cos(el) * cos(az);
    const double nn = sqrt(ex * ex + ey * ey + ez * ez);
    const double zx = -ex / nn, zy = -ey / nn, zz = -ez / nn;
    double xx = zz, xy = 0.0, xz = -zx;                 // x = cross(up, z)
    const double xn = sqrt(xx * xx + xy * xy + xz * xz);
    xx /= xn; xy /= xn; xz /= xn;
    const double yx = zy * xz - zz * xy;                // y = cross(z, x)
    const double yy = zz * xx - zx * xz;
    const double yz = zx * xy - zy * xx;

    vtx_kernel<<<(NVERT + 255) / 256, 256, 0, stream>>>(
        verts, ws, (float)ex, (float)ey, (float)ez,
        (float)xx, (float)xy, (float)xz,
        (float)yx, (float)yy, (float)yz,
        (float)zx, (float)zy, (float)zz);

    face_kernel<<<(NFACE + 255) / 256, 256, 0, stream>>>(faces, ws);

    // 4096 waves (one per 16-pixel tile), 8 waves per block
    raster_kernel<<<NWAVE / 8, 256, 0, stream>>>(ws + WS_FACE, tex, ref, ws + WS_PART);

    reduce_kernel<<<1, 256, 0, stream>>>(ws + WS_PART, (float*)d_out);
}